// ClusteringLayer_14998025798240
// MI455X (gfx1250) — compile-verified
//
#include <hip/hip_runtime.h>
#include <stdint.h>

// Near-duplicate clustering per 64-element cacheline, threshold 0.1.
// Equivalence used: out[j] = out[min i<j with |out[i]-x_j|<T] else x_j,
// realized as a descending select chain (earliest match wins by writing last).

#define THRESH 0.1f
constexpr int LINE      = 64;          // cacheline elements
constexpr int TPB       = 96;          // threads per block (3 waves)
constexpr int LPB_LINES = TPB * 2;     // 192 lines per block, 2 lines per thread
constexpr int STRIDE    = LINE + 2;    // 66 dwords (264B): 8B-aligned pairs;
                                       // 66*96 % 64 == 0 -> both per-thread
                                       // streams use banks {2t+p,2t+p+1}, conflict-free

__global__ __launch_bounds__(TPB)
void cluster_kernel(const float* __restrict__ x, float* __restrict__ out,
                    long long n_lines) {
  __shared__ __align__(16) float lds[LPB_LINES * STRIDE];
  const int tid = threadIdx.x;
  const long long line0 = (long long)blockIdx.x * LPB_LINES;
  long long lh = n_lines - line0;
  if (lh > LPB_LINES) lh = LPB_LINES;
  const int lines_here = (int)lh;
  const int pairs_here = lines_here * (LINE / 2);
  const long long elem0 = line0 * LINE;

  // ---- Stage 1: async DMA global -> LDS (b64 per lane, coalesced) ----
  {
    const float* gsrc = x + elem0;
    for (int c2 = tid; c2 < pairs_here; c2 += TPB) {
      const int c    = c2 * 2;              // even dword index in tile
      const int line = c >> 6;
      const int off  = c & (LINE - 1);      // even -> LDS addr 8B-aligned
      unsigned lds_byte = (unsigned)(uintptr_t)(&lds[line * STRIDE + off]);
      const float* gp = gsrc + c;
      asm volatile("global_load_async_to_lds_b64 %0, %1, off"
                   :: "v"(lds_byte), "v"(gp) : "memory");
    }
    asm volatile("s_wait_asynccnt 0x0" ::: "memory");
  }
  __syncthreads();

  // ---- Stage 2: serial clustering, TWO independent lines per thread ----
  if (tid < lines_here) {
    float* a = &lds[tid * STRIDE];
    // Non-adjacent second line keeps banking conflict-free; if it is out of
    // range, alias it to line A: the B-chain then recomputes A's values and
    // stores identical data to identical addresses (benign, no divergence).
    float* b = (tid + TPB < lines_here) ? &lds[(tid + TPB) * STRIDE] : a;

    for (int j = 1; j < LINE; ++j) {
      const float xa = a[j], xb = b[j];     // position j still holds the input
      float oa = xa, ob = xb;
      int i = j - 1;
      if (j & 1) {                          // peel top single (even index i)
        const float sa = a[i], sb = b[i];
        oa = (fabsf(sa - xa) < THRESH) ? sa : oa;
        ob = (fabsf(sb - xb) < THRESH) ? sb : ob;
        --i;
      }
      // i is now odd: aligned pairs (p, p+1), descending; apply high then low
      #pragma unroll 2
      for (int p = i - 1; p >= 0; p -= 2) {
        const float2 pa = *reinterpret_cast<const float2*>(a + p); // ds_load_b64
        const float2 pb = *reinterpret_cast<const float2*>(b + p); // ds_load_b64
        oa = (fabsf(pa.y - xa) < THRESH) ? pa.y : oa;   // index p+1 first
        ob = (fabsf(pb.y - xb) < THRESH) ? pb.y : ob;
        oa = (fabsf(pa.x - xa) < THRESH) ? pa.x : oa;   // index p wins last
        ob = (fabsf(pb.x - xb) < THRESH) ? pb.x : ob;
      }
      a[j] = oa;
      b[j] = ob;
    }
  }
  __syncthreads();

  // ---- Stage 3: async DMA LDS -> global (b64 per lane, coalesced) ----
  {
    float* gdst = out + elem0;
    for (int c2 = tid; c2 < pairs_here; c2 += TPB) {
      const int c    = c2 * 2;
      const int line = c >> 6;
      const int off  = c & (LINE - 1);
      unsigned lds_byte = (unsigned)(uintptr_t)(&lds[line * STRIDE + off]);
      float* gp = gdst + c;
      asm volatile("global_store_async_from_lds_b64 %0, %1, off"
                   :: "v"(gp), "v"(lds_byte) : "memory");
    }
    asm volatile("s_wait_asynccnt 0x0" ::: "memory");
  }
}

// Elements past the last full cacheline pass through unchanged.
__global__ void tail_copy_kernel(const float* __restrict__ x, float* __restrict__ out,
                                 long long start, long long n) {
  long long i = start + (long long)blockIdx.x * blockDim.x + threadIdx.x;
  if (i < n) out[i] = x[i];
}

extern "C" void kernel_launch(void* const* d_in, const int* in_sizes, int n_in,
                              void* d_out, int out_size, void* d_ws, size_t ws_size,
                              hipStream_t stream) {
  const float* x = (const float*)d_in[0];
  float* out     = (float*)d_out;
  const long long n       = (long long)in_sizes[0];
  const long long n_lines = n / LINE;

  if (n_lines > 0) {
    const int blocks = (int)((n_lines + LPB_LINES - 1) / LPB_LINES);
    cluster_kernel<<<blocks, TPB, 0, stream>>>(x, out, n_lines);
  }
  const long long tstart = n_lines * LINE;
  const long long tail   = n - tstart;
  if (tail > 0) {
    tail_copy_kernel<<<(int)((tail + 255) / 256), 256, 0, stream>>>(x, out, tstart, n);
  }
}